// SimpleRetention_31636729102528
// MI455X (gfx1250) — compile-verified
//
#include <hip/hip_runtime.h>

typedef __attribute__((ext_vector_type(2))) float v2f;
typedef __attribute__((ext_vector_type(8))) float v8f;

#define B_  4
#define L_  4096
#define D_  512
#define HD_ 64
#define VD_ 128

constexpr float LOG2_GAMMA = -0.04580329f;   // log2(0.96875)

__device__ __forceinline__ v8f wmma_f32_16x16x4(v2f a, v2f b, v8f c) {
  // D = A(16x4, f32) * B(4x16, f32) + C(16x16, f32), wave32
  return __builtin_amdgcn_wmma_f32_16x16x4_f32(false, a, false, b, (short)0, c,
                                               false, false);
}

// K-loop over D_=512 for four 16-col tiles of one weight matrix.
// bp already includes (+ colBase + r + 2*hi*STRIDE); all loads are
// immediate-offset off bp, bp advances by a constant -> no per-load addr math.
template <int STRIDE>
__device__ __forceinline__ void gemm4tiles(const float* __restrict__ bp,
                                           const float* __restrict__ xrow,
                                           v8f acc[4]) {
#pragma unroll 4
  for (int k0 = 0; k0 < D_; k0 += 4) {
    v2f a = *(const v2f*)(xrow + k0);   // X[row][k0+2hi], +1
#pragma unroll
    for (int t = 0; t < 4; ++t) {
      v2f b;
      b.x = bp[t * 16];                 // W[k0+2hi  ][c0+t*16+r]
      b.y = bp[t * 16 + STRIDE];        // W[k0+2hi+1][c0+t*16+r]
      acc[t] = wmma_f32_16x16x4(a, b, acc[t]);
    }
    bp += 4 * STRIDE;
  }
}

// --------------------------------------------------------------------------
// Kernel 1: fused projections + xPos. One 16-row tile of X per block (4
// waves); wave w owns 64 consecutive columns of concat [Q(64)|K(64)|V(128)].
// --------------------------------------------------------------------------
__global__ __launch_bounds__(128) void proj_xpos_kernel(
    const float* __restrict__ X,  const float* __restrict__ WQ,
    const float* __restrict__ WK, const float* __restrict__ WV,
    float* __restrict__ Qb, float* __restrict__ Kb, float* __restrict__ Vb) {
  const int lane = threadIdx.x & 31;
  const int w    = threadIdx.x >> 5;   // wave in block: 0..3
  const int r    = lane & 15;          // N column (B/C) / M row (A)
  const int hi   = lane >> 4;          // lane-half -> K sub-pair
  const int rowBase = blockIdx.x * 16;

  const v8f vz = {0.f, 0.f, 0.f, 0.f, 0.f, 0.f, 0.f, 0.f};
  v8f acc[4];
#pragma unroll
  for (int t = 0; t < 4; ++t) acc[t] = vz;

  const float* xrow = X + (size_t)(rowBase + r) * D_ + 2 * hi;
  if (w == 0)      gemm4tiles<HD_>(WQ + r + 2 * hi * HD_, xrow, acc);
  else if (w == 1) gemm4tiles<HD_>(WK + r + 2 * hi * HD_, xrow, acc);
  else             gemm4tiles<VD_>(WV + (w - 2) * 64 + r + 2 * hi * VD_, xrow, acc);

  // Epilogue: C layout = lane holds column N=r, rows M=i+8*hi in acc[t][i].
#pragma unroll
  for (int t = 0; t < 4; ++t) {
    int cc = w * 64 + t * 16 + r;   // global concat column
#pragma unroll
    for (int i = 0; i < 8; ++i) {
      int row = rowBase + i + 8 * hi;
      float x = acc[t][i];
      float p = __shfl_xor(x, 1, 32);   // rotary partner (adjacent column)
      if (cc < 128) {                   // Q or K -> apply xPos
        int  n    = row & (L_ - 1);     // sequence position
        bool down = cc >= 64;           // K uses inverse scale
        int  ch   = cc & 63;
        int  j    = ch >> 1;
        float sv    = (2.0f * (float)j + 0.4f * (float)HD_) * (1.0f / (1.4f * (float)HD_));
        float ex    = (float)n * (1.0f / 512.0f);
        if (down) ex = -ex;
        float scale = exp2f(ex * log2f(sv));          // sv ** (n/512)
        // inv_freq = 10000^(-j/32) = 2^(-j*log2(1e4)/32)
        float invf  = exp2f((float)j * (-13.2877124f / 32.0f));
        float ang   = (float)n * invf;
        float sn, cs;
        sincosf(ang, &sn, &cs);
        sn *= scale; cs *= scale;
        // even col: x*cos - x_odd*sin ; odd col: x*cos + x_even*sin
        float outv = (ch & 1) ? fmaf(x, cs, p * sn) : fmaf(x, cs, -(p * sn));
        if (!down) Qb[(size_t)row * HD_ + ch] = outv;
        else       Kb[(size_t)row * HD_ + ch] = outv;
      } else {
        Vb[(size_t)row * VD_ + (cc - 128)] = x;
      }
    }
  }
}

// --------------------------------------------------------------------------
// Kernel 2: fused retention O = (Q K^T ∘ D) V, flash-style.
// One wave per (batch, 16-query tile); key chunks of 16, truncated at the
// fp32-negligible decay horizon. S transposed C->A layout via padded LDS.
// All chunk-loop global accesses are immediate offsets off loop-carried
// base pointers (kptr / vbase / prefetch ptrs).
// --------------------------------------------------------------------------
__global__ __launch_bounds__(32) void retention_attn_kernel(
    const float* __restrict__ Qb, const float* __restrict__ Kb,
    const float* __restrict__ Vb, float* __restrict__ Ob) {
  __shared__ float sS[16 * 18];   // stride-18: bank-conflict free, f2-aligned
  const int lane = threadIdx.x & 31;
  const int r    = lane & 15;
  const int hi   = lane >> 4;

  const int numQ = L_ / 16;
  const int idx  = blockIdx.x;
  const int bb   = idx / numQ;
  const int qi   = (numQ - 1) - (idx % numQ);   // heavy tiles launch first
  const int n0   = qi * 16;

  const float* Q = Qb + (size_t)bb * L_ * HD_;
  const float* K = Kb + (size_t)bb * L_ * HD_;
  const float* V = Vb + (size_t)bb * L_ * VD_;
  float*       O = Ob + (size_t)bb * L_ * VD_;

  // Q tile as A-frags for all 16 K-steps of hd=64 (kept in registers)
  v2f qa[16];
  {
    const float* qp = Q + (size_t)(n0 + r) * HD_ + 2 * hi;
#pragma unroll
    for (int t = 0; t < 16; ++t) qa[t] = *(const v2f*)(qp + 4 * t);
  }

  // loop-invariant per-lane decay: gamma^(M-N), M=i+8*hi, N=r
  float gp[8];
#pragma unroll
  for (int i = 0; i < 8; ++i)
    gp[i] = exp2f((float)((i + 8 * hi) - r) * LOG2_GAMMA);

  const v8f vz = {0.f, 0.f, 0.f, 0.f, 0.f, 0.f, 0.f, 0.f};
  v8f o[8];
#pragma unroll
  for (int ct = 0; ct < 8; ++ct) o[ct] = vz;

  const int CUT = 1280;                          // gamma^1280 ~ 2^-59
  const int m0s = (n0 > CUT) ? ((n0 - CUT) & ~15) : 0;

  // Loop-carried base pointers (constant increments, immediate offsets)
  const float* kptr  = K + (size_t)(m0s + r) * HD_ + 2 * hi;
  const float* vbase = V + (size_t)m0s * VD_ + 2 * hi * VD_ + r;
  const char*  kpf   = (const char*)(K + (size_t)(m0s + 16 + (lane >> 1)) * HD_)
                       + (lane & 1) * 128;
  const char*  vpf   = (const char*)(V + (size_t)(m0s + 16 + (lane >> 1)) * VD_)
                       + (lane & 1) * 128;

  for (int m0 = m0s; m0 <= n0; m0 += 16) {
    // Prefetch next K/V chunk into cache (gfx1250 global_prefetch_b8)
    if (m0 + 16 <= n0) {
      __builtin_prefetch(kpf, 0, 3);
      __builtin_prefetch(vpf, 0, 3);
      __builtin_prefetch(vpf + 256, 0, 3);
    }

    // S = Q_tile * K_chunk^T  (16 K-steps over hd=64)
    v8f s = vz;
#pragma unroll
    for (int t = 0; t < 16; ++t) {
      v2f kb = *(const v2f*)(kptr + 4 * t);
      s = wmma_f32_16x16x4(qa[t], kb, s);
    }

    // Decay: D[n][m] = gamma^(n0-m0) * gamma^(M-N); causal mask on diagonal.
    if (m0 == n0) {
#pragma unroll
      for (int i = 0; i < 8; ++i) {
        int diff = (i + 8 * hi) - r;
        s[i] = (diff < 0) ? 0.0f : s[i] * gp[i];
      }
    } else {
      float cf = exp2f((float)(n0 - m0) * LOG2_GAMMA);
#pragma unroll
      for (int i = 0; i < 8; ++i) s[i] *= cf * gp[i];
    }

    // Transpose S: C/D layout -> A layout, through LDS (single-wave WG).
    __syncthreads();
#pragma unroll
    for (int i = 0; i < 8; ++i) sS[(i + 8 * hi) * 18 + r] = s[i];
    __syncthreads();
    v2f av[4];
#pragma unroll
    for (int u = 0; u < 4; ++u)
      av[u] = *(const v2f*)&sS[r * 18 + 4 * u + 2 * hi];

    // O += S * V_chunk: all 64 loads immediate offsets off vbase.
#pragma unroll
    for (int u = 0; u < 4; ++u) {
#pragma unroll
      for (int ct = 0; ct < 8; ++ct) {
        v2f vb;
        vb.x = vbase[u * 4 * VD_ + ct * 16];        // V[m0+4u+2hi  ][ct*16+r]
        vb.y = vbase[u * 4 * VD_ + VD_ + ct * 16];  // V[m0+4u+2hi+1][ct*16+r]
        o[ct] = wmma_f32_16x16x4(av[u], vb, o[ct]);
      }
    }

    kptr  += 16 * HD_;
    vbase += 16 * VD_;
    kpf   += 16 * HD_ * sizeof(float);
    vpf   += 16 * VD_ * sizeof(float);
  }

  // Write O (C layout): one base pointer, immediate offsets.
  float* optr = O + (size_t)(n0 + 8 * hi) * VD_ + r;
  float* optr0 = O + (size_t)n0 * VD_ + r;
  (void)optr;
#pragma unroll
  for (int i = 0; i < 8; ++i) {
#pragma unroll
    for (int ct = 0; ct < 8; ++ct) {
      optr0[(size_t)(i + 8 * hi) * VD_ + ct * 16] = o[ct][i];
    }
  }
}

// --------------------------------------------------------------------------
extern "C" void kernel_launch(void* const* d_in, const int* in_sizes, int n_in,
                              void* d_out, int out_size, void* d_ws, size_t ws_size,
                              hipStream_t stream) {
  const float* X  = (const float*)d_in[0];
  const float* WQ = (const float*)d_in[1];
  const float* WK = (const float*)d_in[2];
  const float* WV = (const float*)d_in[3];

  float* Qb = (float*)d_ws;                       // B*L*HD  (4 MB)
  float* Kb = Qb + (size_t)B_ * L_ * HD_;         // B*L*HD  (4 MB)
  float* Vb = Kb + (size_t)B_ * L_ * HD_;         // B*L*VD  (8 MB)
  float* O  = (float*)d_out;

  const int rowTiles = B_ * L_ / 16;              // 1024
  proj_xpos_kernel<<<dim3(rowTiles), dim3(128), 0, stream>>>(X, WQ, WK, WV,
                                                             Qb, Kb, Vb);
  retention_attn_kernel<<<dim3(rowTiles), dim3(32), 0, stream>>>(Qb, Kb, Vb, O);
}